// POSTagger_27736898798242
// MI455X (gfx1250) — compile-verified
//
#include <hip/hip_runtime.h>
#include <hip/hip_bf16.h>
#include <math.h>

// POS tagger: char-CNN -> concat word emb -> GEMM (WMMA fp32, TDM-staged B)
// -> sequential LSTM scan (persistent workgroup, LDS-resident weight slice)
// -> dense + log_softmax.
//
// Dims: S=4096, LC=16, CDIM=10, F=32, WDIM=250, H=250, 4H=1000, K=282->288,
// N=1000->1008, TAGS=50.

#define S_WORDS 4096
#define KP 288        // padded K (282 -> 288)
#define NP 1008       // padded 4H (1000 -> 1008)
#define FOURH 1000
#define H 250
#define TAGS 50
#define KL 76         // K-rows of WhhT pinned in LDS for the scan (304 KB)

typedef float v2f __attribute__((ext_vector_type(2)));
typedef float v8f __attribute__((ext_vector_type(8)));
typedef unsigned int v4u __attribute__((ext_vector_type(4)));
typedef int v8i __attribute__((ext_vector_type(8)));
typedef int v4i __attribute__((ext_vector_type(4)));

// ---------------------------------------------------------------------------
// Kernel 1: char CNN (+max over time) + word-embedding gather -> xpad[S][288]
// One wave per word; lane == filter index (F == 32 == wave32).
// ---------------------------------------------------------------------------
__global__ void __launch_bounds__(128) k_charcnn(
    const int* __restrict__ ci,        // [S*16]
    const int* __restrict__ wi,        // [S]
    const float* __restrict__ cemb,    // [128*10]
    const float* __restrict__ wemb,    // [100000*250]
    const float* __restrict__ cw,      // [32*10*3]
    const float* __restrict__ cb,      // [32]
    float* __restrict__ xpad)          // [S*288]
{
    __shared__ float ce[4][16][10];
    const int lane = threadIdx.x & 31;
    const int w    = threadIdx.x >> 5;
    const int s    = blockIdx.x * 4 + w;

    for (int i = lane; i < 160; i += 32) {
        int t = i / 10, c2 = i % 10;
        int ch = ci[s * 16 + t];
        ce[w][t][c2] = cemb[ch * 10 + c2];
    }
    __syncthreads();

    float wr[30];
#pragma unroll
    for (int i = 0; i < 30; ++i) wr[i] = cw[lane * 30 + i];
    const float bias = cb[lane];

    float best = -3.0e38f;
#pragma unroll
    for (int t = 0; t < 16; ++t) {
        float acc = bias;
#pragma unroll
        for (int c2 = 0; c2 < 10; ++c2) {
            float lft = (t > 0)  ? ce[w][t - 1][c2] : 0.0f;
            float mid = ce[w][t][c2];
            float rgt = (t < 15) ? ce[w][t + 1][c2] : 0.0f;
            acc = fmaf(lft, wr[c2 * 3 + 0], acc);
            acc = fmaf(mid, wr[c2 * 3 + 1], acc);
            acc = fmaf(rgt, wr[c2 * 3 + 2], acc);
        }
        best = fmaxf(best, acc);
    }
    xpad[s * KP + lane] = best;                       // char features [0,32)

    const int widx = wi[s];
    for (int j = lane; j < 250; j += 32)              // word emb [32,282)
        xpad[s * KP + 32 + j] = wemb[widx * 250 + j];
    if (lane < 6)                                     // K padding [282,288)
        xpad[s * KP + 282 + lane] = 0.0f;
}

// ---------------------------------------------------------------------------
// Kernel 2: pack Bpad[288][1008] = W_ih^T (zero padded) and
//           WhhT[250][1024]     = W_hh^T (zero padded)
// ---------------------------------------------------------------------------
__global__ void k_pack(const float* __restrict__ W_ih,   // [1000*282]
                       const float* __restrict__ W_hh,   // [1000*250]
                       float* __restrict__ Bpad,
                       float* __restrict__ WhhT)
{
    const int idx = blockIdx.x * blockDim.x + threadIdx.x;
    const int NB = KP * NP;                // 290304
    if (idx < NB) {
        int k = idx / NP, n = idx % NP;
        Bpad[idx] = (k < 282 && n < FOURH) ? W_ih[n * 282 + k] : 0.0f;
    } else {
        int i2 = idx - NB;
        if (i2 < H * 1024) {
            int k = i2 >> 10, j = i2 & 1023;
            WhhT[i2] = (j < FOURH) ? W_hh[j * H + k] : 0.0f;
        }
    }
}

// ---------------------------------------------------------------------------
// Kernel 3: xg[S][1000] = xpad @ Bpad + (b_ih + b_hh), fp32 WMMA 16x16x4.
// Block = 8 waves; all waves share one N-tile (16 cols). Wave 0 DMAs the
// full 288x16 fp32 B panel (18 KB) into LDS with the Tensor Data Mover
// (tensor_load_to_lds, TENSORcnt), then every wave computes a 16x16 C tile
// with B fragments served from LDS (conflict-free: lane halves 32 banks
// apart). A fragments stream from the L2-resident xpad.
// ---------------------------------------------------------------------------
__global__ void __launch_bounds__(256) k_gemm_xg(
    const float* __restrict__ xpad,   // [S][KP]
    const float* __restrict__ Bpad,   // [KP][NP]
    const float* __restrict__ b_ih,   // [1000]
    const float* __restrict__ b_hh,   // [1000]
    float* __restrict__ xg)           // [S][1000]
{
    __shared__ float lds_b[KP * 16];  // 18 KB: B panel rows k, cols n0..n0+15

    const int tid  = threadIdx.x;
    const int lane = tid & 31;
    const int wave = tid >> 5;
    const int lo   = lane & 15;
    const int hi   = lane >> 4;
    const int n0   = blockIdx.x * 16;                 // 63 tiles -> [0,1008)
    const int m0   = (blockIdx.y * 8 + wave) * 16;    // 32*8 tiles -> [0,4096)

    if (wave == 0) {
        // Tensor DMA descriptor (D#), 2D tile: 16 elems wide x 288 rows,
        // 4-byte elements, row stride 1008 elems.
        unsigned long long ga = (unsigned long long)(uintptr_t)(Bpad + n0);
        unsigned int lb = (unsigned int)(uintptr_t)(&lds_b[0]);
        v4u g0 = { 1u,                                   // count=1, user mode
                   lb,                                   // lds_addr
                   (unsigned int)(ga & 0xffffffffull),   // global_addr[31:0]
                   (unsigned int)(((ga >> 32) & 0x01ffffffull) | 0x80000000u) };
                                                         // addr[56:32] | type=2
        v8i g1 = { (int)0x20000,          // data_size=2 (4B), no mask/pad
                   (int)(1008u << 16),    // tensor_dim0[15:0] << 16
                   (int)(288u << 16),     // dim0 hi=0 | tensor_dim1[15:0]<<16
                   (int)(16u << 16),      // dim1 hi=0 | tile_dim0=16
                   288,                   // tile_dim1=288 | tile_dim2=0
                   1008,                  // tensor_dim0_stride[31:0]
                   0, 0 };                // stride0 hi | stride1 (unused, 2D)
        v4i gz = { 0, 0, 0, 0 };
        v8i gz8 = { 0, 0, 0, 0, 0, 0, 0, 0 };
        __builtin_amdgcn_tensor_load_to_lds(g0, g1, gz, gz, gz8, 0);
        __builtin_amdgcn_s_wait_tensorcnt(0);
    }
    __syncthreads();

    v8f c = {};
    const float* arow = xpad + (m0 + lo) * KP + 2 * hi;
    const float* brow = lds_b + (2 * hi) * 16 + lo;

#pragma unroll 4
    for (int k0 = 0; k0 < KP; k0 += 4) {
        v2f a = *(const v2f*)(arow + k0);   // A[M][k0+2*hi .. +1], 8B aligned
        v2f b;
        b.x = brow[k0 * 16];                // B[k0+2*hi+0][N]
        b.y = brow[k0 * 16 + 16];           // B[k0+2*hi+1][N]
        c = __builtin_amdgcn_wmma_f32_16x16x4_f32(
                /*neg_a=*/false, a, /*neg_b=*/false, b,
                /*c_mod=*/(short)0, c, /*reuse_a=*/false, /*reuse_b=*/false);
    }

    const int n = n0 + lo;
    if (n < FOURH) {
        const float bias = b_ih[n] + b_hh[n];
#pragma unroll
        for (int v = 0; v < 8; ++v) {
            const int m = m0 + v + 8 * hi;
            xg[m * FOURH + n] = c[v] + bias;
        }
    }
}

// ---------------------------------------------------------------------------
// Kernel 4: sequential LSTM scan. Single persistent workgroup (1024 threads
// = 32 wave32s on one WGP). h/c/gates in LDS. The first KL=76 K-rows of
// WhhT (304 KB) are pinned in the WGP's 320 KB LDS; the remaining 174 rows
// stream coalesced from the L2-resident transposed weight each step.
// ---------------------------------------------------------------------------
__global__ void __launch_bounds__(1024) k_lstm_scan(
    const float* __restrict__ xg,     // [S][1000]
    const float* __restrict__ WhhT,   // [250][1024], cols >= 1000 zero
    float* __restrict__ hs)           // [S][250]
{
    __shared__ float wlds[KL << 10];  // 76 * 1024 floats = 304 KB
    __shared__ float h_s[256];
    __shared__ float c_s[256];
    __shared__ float g_s[1024];
    const int tid = threadIdx.x;

    for (int i = tid; i < (KL << 10); i += 1024) wlds[i] = WhhT[i];
    if (tid < 256) { h_s[tid] = 0.0f; c_s[tid] = 0.0f; }
    __syncthreads();

    for (int t = 0; t < S_WORDS; ++t) {
        float acc = (tid < FOURH) ? xg[t * FOURH + tid] : 0.0f;
#pragma unroll 4
        for (int k = 0; k < KL; ++k)              // LDS-resident slice
            acc = fmaf(wlds[(k << 10) + tid], h_s[k], acc);
#pragma unroll 6
        for (int k = KL; k < H; ++k)              // L2-streamed remainder
            acc = fmaf(WhhT[(k << 10) + tid], h_s[k], acc);
        g_s[tid] = acc;
        if (t + 1 < S_WORDS && tid < FOURH)
            __builtin_prefetch(xg + (t + 1) * FOURH + tid, 0, 0);
        __syncthreads();

        if (tid < H) {
            const float ig = 1.0f / (1.0f + expf(-g_s[tid]));
            const float fg = 1.0f / (1.0f + expf(-g_s[H + tid]));
            const float gg = tanhf(g_s[2 * H + tid]);
            const float og = 1.0f / (1.0f + expf(-g_s[3 * H + tid]));
            const float cc = fmaf(fg, c_s[tid], ig * gg);
            c_s[tid] = cc;
            const float hh = og * tanhf(cc);
            h_s[tid] = hh;
            hs[t * H + tid] = hh;
        }
        __syncthreads();
    }
}

// ---------------------------------------------------------------------------
// Kernel 5: dense + log_softmax. One 64-thread block per word.
// ---------------------------------------------------------------------------
__global__ void __launch_bounds__(64) k_dense_lsm(
    const float* __restrict__ hs,       // [S][250]
    const float* __restrict__ dense_w,  // [50*250]
    const float* __restrict__ dense_b,  // [50]
    float* __restrict__ out)            // [S][50]
{
    __shared__ float red[64];
    const int s = blockIdx.x, tid = threadIdx.x;

    float logit = 0.0f;
    float mval = -3.0e38f;
    if (tid < TAGS) {
        logit = dense_b[tid];
        const float* wrow = dense_w + tid * H;
        const float* hrow = hs + s * H;
#pragma unroll 5
        for (int k = 0; k < H; ++k) logit = fmaf(wrow[k], hrow[k], logit);
        mval = logit;
    }
    red[tid] = mval;
    __syncthreads();
    for (int off = 32; off > 0; off >>= 1) {
        if (tid < off) red[tid] = fmaxf(red[tid], red[tid + off]);
        __syncthreads();
    }
    const float mx = red[0];
    __syncthreads();
    red[tid] = (tid < TAGS) ? expf(logit - mx) : 0.0f;
    __syncthreads();
    for (int off = 32; off > 0; off >>= 1) {
        if (tid < off) red[tid] += red[tid + off];
        __syncthreads();
    }
    const float lse = logf(red[0]);
    if (tid < TAGS) out[s * TAGS + tid] = logit - mx - lse;
}

// ---------------------------------------------------------------------------
// Launch. Workspace layout (256B-aligned, ~22.3 MB total):
//   [0,        4718592)  xpad [4096*288] f32   (reused as hs later)
//   [4718592,  5879808)  Bpad [288*1008] f32
//   [5879808, 22263808)  xg   [4096*1000] f32
//   [22263808,23287808)  WhhT [250*1024] f32
// hs aliases xpad: xpad is dead once k_gemm_xg finishes (stream-ordered).
// ---------------------------------------------------------------------------
extern "C" void kernel_launch(void* const* d_in, const int* in_sizes, int n_in,
                              void* d_out, int out_size, void* d_ws, size_t ws_size,
                              hipStream_t stream) {
    const int*   ci      = (const int*)d_in[0];
    const int*   wi      = (const int*)d_in[1];
    const float* cemb    = (const float*)d_in[2];
    const float* wemb    = (const float*)d_in[3];
    const float* cw      = (const float*)d_in[4];
    const float* cb      = (const float*)d_in[5];
    const float* W_ih    = (const float*)d_in[6];
    const float* W_hh    = (const float*)d_in[7];
    const float* b_ih    = (const float*)d_in[8];
    const float* b_hh    = (const float*)d_in[9];
    const float* dense_w = (const float*)d_in[10];
    const float* dense_b = (const float*)d_in[11];
    float* out = (float*)d_out;

    char* ws = (char*)d_ws;
    float* xpad = (float*)(ws);
    float* Bpad = (float*)(ws + 4718592);
    float* xg   = (float*)(ws + 5879808);
    float* WhhT = (float*)(ws + 22263808);
    float* hs   = (float*)(ws);           // alias over dead xpad

    k_charcnn<<<dim3(S_WORDS / 4), dim3(128), 0, stream>>>(
        ci, wi, cemb, wemb, cw, cb, xpad);

    {
        const int total = KP * NP + H * 1024;
        k_pack<<<dim3((total + 255) / 256), dim3(256), 0, stream>>>(
            W_ih, W_hh, Bpad, WhhT);
    }

    k_gemm_xg<<<dim3(NP / 16, S_WORDS / 16 / 8), dim3(256), 0, stream>>>(
        xpad, Bpad, b_ih, b_hh, xg);

    k_lstm_scan<<<dim3(1), dim3(1024), 0, stream>>>(xg, WhhT, hs);

    k_dense_lsm<<<dim3(S_WORDS), dim3(64), 0, stream>>>(
        hs, dense_w, dense_b, out);
}